// ClefDecoder_63522566308361
// MI455X (gfx1250) — compile-verified
//
#include <hip/hip_runtime.h>
#include <cstdint>
#include <cstddef>

typedef __attribute__((ext_vector_type(2))) float v2f;
typedef __attribute__((ext_vector_type(8))) float v8f;

constexpr int B_ = 8, S_ = 4096, DN_ = 256;
constexpr int MROWS = B_ * S_;       // 32768 token rows
constexpr int G3 = 3 * DN_;          // 768

__device__ __forceinline__ v8f wmma4(v2f a, v2f b, v8f c) {
  // V_WMMA_F32_16X16X4_F32 : D = A(16x4,f32) * B(4x16,f32) + C(16x16,f32)
  return __builtin_amdgcn_wmma_f32_16x16x4_f32(false, a, false, b, (short)0, c,
                                               false, false);
}

__device__ __forceinline__ v8f vzero8() {
  v8f v = {0.f, 0.f, 0.f, 0.f, 0.f, 0.f, 0.f, 0.f};
  return v;
}

__device__ __forceinline__ float sigmoidf_(float x) {
  return 1.f / (1.f + __expf(-x));
}

// -------------------------------------------------------------------------
// Generic transpose: out[c*R + r] = in[r*C + c]   (in: [R][C], out: [C][R])
// -------------------------------------------------------------------------
__global__ void transpose_k(const float* __restrict__ in, float* __restrict__ out,
                            int R, int C) {
  int idx = blockIdx.x * blockDim.x + threadIdx.x;
  if (idx >= R * C) return;
  int r = idx / C, c = idx % C;
  out[(size_t)c * R + r] = in[idx];
}

// -------------------------------------------------------------------------
// Segment builder: one thread per batch scans bar_raw; segments end at bars.
// -------------------------------------------------------------------------
__global__ void build_segments(const int* __restrict__ bar,
                               int* __restrict__ seg_start,
                               int* __restrict__ seg_len,
                               int* __restrict__ nseg) {
  int b = threadIdx.x;
  if (b >= B_) return;
  const int* br = bar + (size_t)b * S_;
  int cnt = 0, cur = 0;
  for (int s = 0; s < S_; ++s) {
    if (br[s] == 0) {
      seg_start[b * S_ + cnt] = cur;
      seg_len[b * S_ + cnt] = s - cur + 1;
      ++cnt;
      cur = s + 1;
    }
  }
  if (cur < S_) {
    seg_start[b * S_ + cnt] = cur;
    seg_len[b * S_ + cnt] = S_ - cur;
    ++cnt;
  }
  nseg[b] = cnt;
}

// -------------------------------------------------------------------------
// WMMA fp32 GEMM: C[M,N] = A[M,K] @ B + bias. A row-major (lda), B element
// (k,n) at Bm[k*bsk + n*bsn]. Block = 256 thr = 8 waves. Block tile 64x128;
// each wave owns a 2x2 grid of 16x16 WMMA tiles (A/B fragment reuse x2).
// Requires M%64==0, N%128==0, K%32==0 (true for all three calls).
// -------------------------------------------------------------------------
__global__ __launch_bounds__(256) void gemm_wmma(
    const float* __restrict__ A, int lda, const float* __restrict__ Bm, int bsk,
    int bsn, const float* __restrict__ bias, float* __restrict__ C, int N,
    int K) {
  __shared__ __align__(16) float As[64][34];
  __shared__ __align__(16) float Bs[128][34];
  const int tid = threadIdx.x;
  const int mBase = blockIdx.y * 64;
  const int nBase = blockIdx.x * 128;
  const int wave = tid >> 5, lane = tid & 31;
  const int l15 = lane & 15, hi = lane >> 4;
  const int m0 = (wave >> 2) * 32;   // 0 or 32
  const int n0 = (wave & 3) * 32;    // 0,32,64,96
  v8f acc[2][2];
#pragma unroll
  for (int mt = 0; mt < 2; ++mt)
#pragma unroll
    for (int nt = 0; nt < 2; ++nt) acc[mt][nt] = vzero8();

  for (int kc = 0; kc < K; kc += 32) {
    // stage A: 64x32
    for (int i = tid; i < 64 * 32; i += 256) {
      int m = i >> 5, k = i & 31;
      As[m][k] = A[(size_t)(mBase + m) * lda + kc + k];
    }
    // stage B: 32x128, stored n-major
    for (int i = tid; i < 32 * 128; i += 256) {
      int k = i >> 7, n = i & 127;
      Bs[n][k] = Bm[(size_t)(kc + k) * bsk + (size_t)(nBase + n) * bsn];
    }
    // prefetch next chunk into cache while this one is consumed
    if (kc + 32 < K) {
      if (tid < 64)
        __builtin_prefetch(&A[(size_t)(mBase + tid) * lda + kc + 32], 0, 3);
      if (tid < 128) {
        int k = tid >> 2, seg = tid & 3;
        __builtin_prefetch(
            &Bm[(size_t)(kc + 32 + k) * bsk + (size_t)(nBase + seg * 32) * bsn],
            0, 3);
      }
    }
    __syncthreads();
#pragma unroll
    for (int k0 = 0; k0 < 32; k0 += 4) {
      v2f av0 = *(const v2f*)&As[m0 + l15][k0 + 2 * hi];
      v2f av1 = *(const v2f*)&As[m0 + 16 + l15][k0 + 2 * hi];
      v2f bv0 = *(const v2f*)&Bs[n0 + l15][k0 + 2 * hi];
      v2f bv1 = *(const v2f*)&Bs[n0 + 16 + l15][k0 + 2 * hi];
      acc[0][0] = wmma4(av0, bv0, acc[0][0]);
      acc[0][1] = wmma4(av0, bv1, acc[0][1]);
      acc[1][0] = wmma4(av1, bv0, acc[1][0]);
      acc[1][1] = wmma4(av1, bv1, acc[1][1]);
    }
    __syncthreads();
  }
#pragma unroll
  for (int mt = 0; mt < 2; ++mt) {
#pragma unroll
    for (int nt = 0; nt < 2; ++nt) {
#pragma unroll
      for (int jv = 0; jv < 8; ++jv) {
        int row = mBase + m0 + mt * 16 + jv + 8 * hi;
        int col = nBase + n0 + nt * 16 + l15;
        C[(size_t)row * N + col] = acc[mt][nt][jv] + bias[col];
      }
    }
  }
}

// -------------------------------------------------------------------------
// Segment-parallel GRU scan. Block = group of 16 independent segments of one
// batch (bar tokens reset the recurrence, so segments are independent),
// 256 threads = 8 waves. H[16][256] in LDS. Per step:
//   phase 1: time_center dot (h_before) + HG = H @ Whh^T via WMMA; wave w
//            owns hidden cols [32w,32w+32) of all 3 gates (6 acc tiles), so
//            the gate nonlinearity is pure register math.
//   phase 2: gates, h_new, H update, output stores.
// -------------------------------------------------------------------------
__global__ __launch_bounds__(256) void gru_scan(
    const float* __restrict__ xg,       // [B,S,768]
    const float* __restrict__ resetv,   // [B,S,256]
    const float* __restrict__ wht,      // W_hh^T, [256][768]
    const float* __restrict__ bhh,      // [768]
    const float* __restrict__ wtime,    // [256]
    const float* __restrict__ btime,    // [1]
    const float* __restrict__ comt,     // [B,S]
    const int* __restrict__ bar,        // [B,S]
    const int* __restrict__ seg_start, const int* __restrict__ seg_len,
    const int* __restrict__ nsegArr, float* __restrict__ out /* [B,S,257] */) {
  __shared__ __align__(16) float Hs[16][258];
  __shared__ int s_start[16], s_len[16], s_maxlen;

  const int b = blockIdx.y;
  const int g = blockIdx.x;
  const int nseg = nsegArr[b];
  const int sg0 = g * 16;
  if (sg0 >= nseg) return;

  const int tid = threadIdx.x;
  const int wave = tid >> 5, lane = tid & 31;
  const int l15 = lane & 15, hi = lane >> 4;
  const int jb = wave * 32;  // this wave's hidden-column base

  if (tid < 16) {
    int sidx = sg0 + tid;
    if (sidx < nseg) {
      s_start[tid] = seg_start[b * S_ + sidx];
      s_len[tid] = seg_len[b * S_ + sidx];
    } else {
      s_start[tid] = 0;
      s_len[tid] = 0;
    }
  }
  __syncthreads();
  if (tid == 0) {
    int m = 0;
    for (int r = 0; r < 16; ++r) m = max(m, s_len[r]);
    s_maxlen = m;
  }
  for (int i = tid; i < 16 * 256; i += 256) {
    int r = i >> 8, j = i & 255;
    float h0 = 0.f;
    if (s_start[r] > 0 && s_len[r] > 0)
      h0 = resetv[((size_t)b * S_ + (s_start[r] - 1)) * (size_t)DN_ + j];
    Hs[r][j] = h0;
  }
  __syncthreads();
  const int maxlen = s_maxlen;
  const float bt = btime[0];

  for (int t = 0; t < maxlen; ++t) {
    // ---- phase 1a: time_center = sigmoid(h_before . W_time + b_time) ----
    {
      int r = tid >> 4, q = tid & 15;
      const bool ract = ((sg0 + r) < nseg) && (t < s_len[r]);
      if (ract) {
        // prefetch this row's xg gates for phase 2 (16 thr x 192B = 3KB row)
        const float* xrow =
            xg + ((size_t)b * S_ + (s_start[r] + t)) * (size_t)G3;
        __builtin_prefetch(xrow + q * 48, 0, 3);
      }
      float pd = 0.f;
      for (int j = q; j < 256; j += 16) pd += Hs[r][j] * wtime[j];
      pd += __shfl_xor(pd, 1);
      pd += __shfl_xor(pd, 2);
      pd += __shfl_xor(pd, 4);
      pd += __shfl_xor(pd, 8);
      if (q == 0 && ract) {
        int s = s_start[r] + t;
        float tc = (bar[(size_t)b * S_ + s] == 0)
                       ? comt[(size_t)b * S_ + s]
                       : sigmoidf_(pd + bt);
        out[((size_t)b * S_ + s) * 257] = tc;
      }
    }
    // ---- phase 1b: HG tiles via WMMA (A = H rows, B = Whh^T columns) ----
    v8f acc[3][2];
#pragma unroll
    for (int gi = 0; gi < 3; ++gi)
#pragma unroll
      for (int ct = 0; ct < 2; ++ct) acc[gi][ct] = vzero8();

#pragma unroll 2
    for (int k0 = 0; k0 < 256; k0 += 4) {
      v2f av = *(const v2f*)&Hs[l15][k0 + 2 * hi];
      const float* w0 = wht + (size_t)(k0 + 2 * hi) * G3;
#pragma unroll
      for (int gi = 0; gi < 3; ++gi) {
#pragma unroll
        for (int ct = 0; ct < 2; ++ct) {
          int n = gi * 256 + jb + ct * 16 + l15;
          v2f bv = {w0[n], w0[G3 + n]};
          acc[gi][ct] = wmma4(av, bv, acc[gi][ct]);
        }
      }
    }
    __syncthreads();  // all H reads done before H is rewritten

    // ---- phase 2: gates + state update + h_after stores ----
#pragma unroll
    for (int ct = 0; ct < 2; ++ct) {
      const int j = jb + ct * 16 + l15;
      const float bhr = bhh[j], bhz = bhh[256 + j], bhn = bhh[512 + j];
#pragma unroll
      for (int jv = 0; jv < 8; ++jv) {
        const int m = jv + 8 * hi;  // row (segment slot) in WMMA C layout
        const bool act = ((sg0 + m) < nseg) && (t < s_len[m]);
        const int s = act ? (s_start[m] + t) : 0;
        const size_t xbase = ((size_t)b * S_ + s) * (size_t)G3;
        float xr = xg[xbase + j];
        float xz = xg[xbase + 256 + j];
        float xn = xg[xbase + 512 + j];
        float rg = sigmoidf_(xr + acc[0][ct][jv] + bhr);
        float zg = sigmoidf_(xz + acc[1][ct][jv] + bhz);
        float ng = tanhf(xn + rg * (acc[2][ct][jv] + bhn));
        float hold = Hs[m][j];
        float hnew = (1.f - zg) * ng + zg * hold;
        if (act) {
          out[((size_t)b * S_ + s) * 257 + 1 + j] = hnew;
          Hs[m][j] = hnew;
        }
      }
    }
    __syncthreads();
  }
}

// -------------------------------------------------------------------------
extern "C" void kernel_launch(void* const* d_in, const int* in_sizes, int n_in,
                              void* d_out, int out_size, void* d_ws,
                              size_t ws_size, hipStream_t stream) {
  (void)in_sizes; (void)n_in; (void)out_size; (void)ws_size;
  const float* tgt    = (const float*)d_in[0];
  const float* hbar   = (const float*)d_in[1];
  const float* comt   = (const float*)d_in[2];
  const float* W_in   = (const float*)d_in[3];
  const float* b_in   = (const float*)d_in[4];
  const float* W_init = (const float*)d_in[5];
  const float* b_init = (const float*)d_in[6];
  const float* W_ih   = (const float*)d_in[7];
  const float* b_ih   = (const float*)d_in[8];
  const float* W_hh   = (const float*)d_in[9];
  const float* b_hh   = (const float*)d_in[10];
  const float* W_time = (const float*)d_in[11];
  const float* b_time = (const float*)d_in[12];
  const int*   bar    = (const int*)d_in[13];
  float* out = (float*)d_out;

  char* ws = (char*)d_ws;
  float* XG = (float*)ws;        ws += (size_t)MROWS * G3 * sizeof(float);
  float* Xbuf = (float*)ws;      ws += (size_t)MROWS * DN_ * sizeof(float);
  float* WIT = (float*)ws;       ws += (size_t)G3 * DN_ * sizeof(float);
  float* WHT = (float*)ws;       ws += (size_t)G3 * DN_ * sizeof(float);
  int* segS = (int*)ws;          ws += (size_t)B_ * S_ * sizeof(int);
  int* segL = (int*)ws;          ws += (size_t)B_ * S_ * sizeof(int);
  int* nseg = (int*)ws;          ws += 64;

  int tot = G3 * DN_;
  transpose_k<<<(tot + 255) / 256, 256, 0, stream>>>(W_ih, WIT, G3, DN_);
  transpose_k<<<(tot + 255) / 256, 256, 0, stream>>>(W_hh, WHT, G3, DN_);
  build_segments<<<1, 32, 0, stream>>>(bar, segS, segL, nseg);

  // X = tgt @ W_in + b_in        [32768,512] x [512,256]
  gemm_wmma<<<dim3(DN_ / 128, MROWS / 64), 256, 0, stream>>>(
      tgt, 512, W_in, DN_, 1, b_in, Xbuf, DN_, 512);
  // XG = X @ W_ih^T + b_ih       [32768,256] x [256,768]  (B = WIT)
  gemm_wmma<<<dim3(G3 / 128, MROWS / 64), 256, 0, stream>>>(
      Xbuf, DN_, WIT, G3, 1, b_ih, XG, G3, DN_);
  // RESET = hbar @ W_init + b_init  (reuses Xbuf; only bar rows consumed)
  gemm_wmma<<<dim3(DN_ / 128, MROWS / 64), 256, 0, stream>>>(
      hbar, DN_, W_init, DN_, 1, b_init, Xbuf, DN_, DN_);

  gru_scan<<<dim3(S_ / 16, B_), 256, 0, stream>>>(
      XG, Xbuf, WHT, b_hh, W_time, b_time, comt, bar, segS, segL, nseg, out);
}